// DeformConvWithChangeableStride_39779987095718
// MI455X (gfx1250) — compile-verified
//
#include <hip/hip_runtime.h>
#include <hip/hip_bf16.h>

typedef float v2f __attribute__((ext_vector_type(2)));
typedef float v8f __attribute__((ext_vector_type(8)));

#define N_IMG   8
#define C_IN    256
#define O_OUT   256
#define H_IMG   64
#define W_IMG   64
#define HW      (H_IMG * W_IMG)      // 4096
#define KS      9                    // 3x3
#define KTOT    (C_IN * KS)          // 2304  (GEMM K)
#define CB      32                   // channels per K-chunk
#define KC      (CB * KS)            // 288
#define NPIX    32                   // pixels per block (GEMM N tile)
#define SSTRIDE (KC + 2)             // 290: even (8B-aligned b64) + bank spread
#define PADDING 1

__global__ __launch_bounds__(256)
void deform_conv_wmma_kernel(const float* __restrict__ x,
                             const float* __restrict__ offset,
                             const float* __restrict__ weight,
                             const float* __restrict__ bias,
                             const int*   __restrict__ stride_p,
                             float*       __restrict__ out)
{
    __shared__ float S[NPIX * SSTRIDE];        // sampled tile: [pixel][c_local*9+k]
    __shared__ int   cIdx[4][KS * NPIX];       // clamped corner linear indices
    __shared__ float cWgt[4][KS * NPIX];       // validity-masked bilinear weights

    const int tid = threadIdx.x;
    const int blk = blockIdx.x;
    const int n   = blk >> 7;                  // 128 blocks per image
    const int p0  = (blk & 127) * NPIX;        // linear pixel base within image
    const int h   = p0 >> 6;                   // output row
    const int w0  = p0 & 63;                   // output col base (0 or 32)
    const int stride = stride_p[0];            // =1 in this setup

    // ---------------- corner precompute: depends on (k, pixel) only ----------------
    for (int idx = tid; idx < KS * NPIX; idx += 256) {
        const int k = idx >> 5;                // 0..8
        const int p = idx & 31;
        const float dy = offset[(((size_t)n * 2 * KS + 2 * k    ) * H_IMG + h) * W_IMG + w0 + p];
        const float dx = offset[(((size_t)n * 2 * KS + 2 * k + 1) * H_IMG + h) * W_IMG + w0 + p];
        const float py = dy + (float)(k / 3) + (float)(h * stride - PADDING);
        const float px = dx + (float)(k % 3) + (float)((w0 + p) * stride - PADDING);
        const float fy = floorf(py), fx = floorf(px);
        const float ly = py - fy,   lx = px - fx;
        const int   y0 = (int)fy,   x0 = (int)fx;
        const float wts[4] = { (1.f - ly) * (1.f - lx), (1.f - ly) * lx,
                               ly * (1.f - lx),          ly * lx };
#pragma unroll
        for (int c4 = 0; c4 < 4; ++c4) {
            const int yi = y0 + (c4 >> 1);
            const int xi = x0 + (c4 & 1);
            const bool valid = (yi >= 0) && (yi < H_IMG) && (xi >= 0) && (xi < W_IMG);
            const int cy = min(max(yi, 0), H_IMG - 1);
            const int cx = min(max(xi, 0), W_IMG - 1);
            cIdx[c4][idx] = cy * W_IMG + cx;
            cWgt[c4][idx] = valid ? wts[c4] : 0.f;
        }
    }

    const int lane = tid & 31;
    const int wv   = tid >> 5;                 // wave 0..7
    const int m0   = wv * 32;                  // 32 output channels per wave
    const int l15  = lane & 15;
    const int koff = (lane >> 4) * 2;          // fp32 WMMA A/B lane K-offset (0 or 2)

    v8f acc[2][2];
#pragma unroll
    for (int a = 0; a < 2; ++a)
#pragma unroll
        for (int b = 0; b < 2; ++b)
#pragma unroll
            for (int e = 0; e < 8; ++e) acc[a][b][e] = 0.f;

    const float* __restrict__ xn = x + (size_t)n * C_IN * HW;
    __syncthreads();

    // ---------------- K-chunk loop: build S tile, then WMMA GEMM ----------------
    for (int chunk = 0; chunk < C_IN / CB; ++chunk) {
        const int c0 = chunk * CB;

        // Phase 1: fused bilinear im2col into LDS (9216 samples / 256 threads)
        for (int idx = tid; idx < CB * KS * NPIX; idx += 256) {
            const int cl = idx / (KS * NPIX);
            const int kp = idx - cl * (KS * NPIX);
            const float* __restrict__ xc = xn + (size_t)(c0 + cl) * HW;
            const float v = xc[cIdx[0][kp]] * cWgt[0][kp]
                          + xc[cIdx[1][kp]] * cWgt[1][kp]
                          + xc[cIdx[2][kp]] * cWgt[2][kp]
                          + xc[cIdx[3][kp]] * cWgt[3][kp];
            const int k = kp >> 5;
            const int p = kp & 31;
            S[p * SSTRIDE + cl * KS + k] = v;
        }
        __syncthreads();

        // Phase 2: 72 steps of v_wmma_f32_16x16x4_f32 per accumulator
        const int ck0 = c0 * KS;               // chunk * 288 (even)
        const float* __restrict__ wA0 = weight + (size_t)(m0 +      l15) * KTOT + ck0 + koff;
        const float* __restrict__ wA1 = weight + (size_t)(m0 + 16 + l15) * KTOT + ck0 + koff;
        const float* __restrict__ sB0 = S + (l15     ) * SSTRIDE + koff;
        const float* __restrict__ sB1 = S + (l15 + 16) * SSTRIDE + koff;

#pragma unroll 4
        for (int kk = 0; kk < KC; kk += 4) {
            const v2f a0 = *(const v2f*)(wA0 + kk);   // W rows m0..m0+15,   K=kk..kk+3
            const v2f a1 = *(const v2f*)(wA1 + kk);   // W rows m0+16..+31
            const v2f b0 = *(const v2f*)(sB0 + kk);   // S cols 0..15  (pixels)
            const v2f b1 = *(const v2f*)(sB1 + kk);   // S cols 16..31
            acc[0][0] = __builtin_amdgcn_wmma_f32_16x16x4_f32(false, a0, false, b0, (short)0, acc[0][0], false, false);
            acc[0][1] = __builtin_amdgcn_wmma_f32_16x16x4_f32(false, a0, false, b1, (short)0, acc[0][1], false, false);
            acc[1][0] = __builtin_amdgcn_wmma_f32_16x16x4_f32(false, a1, false, b0, (short)0, acc[1][0], false, false);
            acc[1][1] = __builtin_amdgcn_wmma_f32_16x16x4_f32(false, a1, false, b1, (short)0, acc[1][1], false, false);
        }
        __syncthreads();
    }

    // ---------------- D-matrix store: VGPR r <-> M=r (+8 for lanes 16..31), N=lane&15 ----------------
    const int mhi = (lane >> 4) * 8;
#pragma unroll
    for (int mi = 0; mi < 2; ++mi)
#pragma unroll
        for (int ni = 0; ni < 2; ++ni)
#pragma unroll
            for (int r = 0; r < 8; ++r) {
                const int o   = m0 + mi * 16 + mhi + r;
                const int col = p0 + ni * 16 + l15;
                out[((size_t)n * O_OUT + o) * HW + col] = acc[mi][ni][r] + bias[o];
            }
}

extern "C" void kernel_launch(void* const* d_in, const int* in_sizes, int n_in,
                              void* d_out, int out_size, void* d_ws, size_t ws_size,
                              hipStream_t stream) {
    const float* x      = (const float*)d_in[0];   // [8,256,64,64]
    const float* offset = (const float*)d_in[1];   // [8,18,64,64]
    const float* weight = (const float*)d_in[2];   // [256,256,3,3]
    const float* bias   = (const float*)d_in[3];   // [256]
    const int*   stride = (const int*)d_in[4];     // scalar (=1)
    float*       out    = (float*)d_out;           // [8,256,64,64]

    const dim3 grid(N_IMG * (HW / NPIX));          // 1024 blocks
    deform_conv_wmma_kernel<<<grid, 256, 0, stream>>>(x, offset, weight, bias, stride, out);
}